// FinalModel_1443109012144
// MI455X (gfx1250) — compile-verified
//
#include <hip/hip_runtime.h>
#include <hip/hip_bf16.h>
#include <stdint.h>

// Top-k gating (k=8, E=64) for MI455X / gfx1250.
// Bandwidth-bound: 512 MB traffic -> ~22 us at 23.3 TB/s. The CDNA5 async
// global<->LDS DMA path (ASYNCcnt) provides fully coalesced 512B-per-issue
// transfers; a Batcher selection network keeps VALU well under the HBM time.

#define BLOCK     128
#define WAVES     (BLOCK / 32)
#define NEXPERT   64
#define ROWSTRIDE 68          // floats; 272 B pitch: 16B-aligned rows, 2-way-max bank conflicts
#define L1_EPS    1e-12f

// Compare-and-swap, descending (max to first slot).
__device__ __forceinline__ void cas(float& x, float& y) {
  float h = fmaxf(x, y);
  y = fminf(x, y);
  x = h;
}

// Batcher odd-even mergesort, 8 elements, descending. 19 comparators.
__device__ __forceinline__ void sort8(float (&a)[8]) {
  // sort4 on [0..3]
  cas(a[0], a[1]); cas(a[2], a[3]); cas(a[0], a[2]); cas(a[1], a[3]); cas(a[1], a[2]);
  // sort4 on [4..7]
  cas(a[4], a[5]); cas(a[6], a[7]); cas(a[4], a[6]); cas(a[5], a[7]); cas(a[5], a[6]);
  // odd-even merge of the two sorted 4-lists
  cas(a[0], a[4]); cas(a[1], a[5]); cas(a[2], a[6]); cas(a[3], a[7]);
  cas(a[2], a[4]); cas(a[3], a[5]);
  cas(a[1], a[2]); cas(a[3], a[4]); cas(a[5], a[6]);
}

// a, b sorted descending. a <- top-8 of (a ∪ b), sorted descending.
// Half-cleaner (8 fmax) leaves a bitonic; 12-comparator bitonic merge re-sorts.
__device__ __forceinline__ void merge_top8(float (&a)[8], const float (&b)[8]) {
  a[0] = fmaxf(a[0], b[7]); a[1] = fmaxf(a[1], b[6]);
  a[2] = fmaxf(a[2], b[5]); a[3] = fmaxf(a[3], b[4]);
  a[4] = fmaxf(a[4], b[3]); a[5] = fmaxf(a[5], b[2]);
  a[6] = fmaxf(a[6], b[1]); a[7] = fmaxf(a[7], b[0]);
  cas(a[0], a[4]); cas(a[1], a[5]); cas(a[2], a[6]); cas(a[3], a[7]);
  cas(a[0], a[2]); cas(a[1], a[3]); cas(a[4], a[6]); cas(a[5], a[7]);
  cas(a[0], a[1]); cas(a[2], a[3]); cas(a[4], a[5]); cas(a[6], a[7]);
}

__global__ __launch_bounds__(BLOCK) void topk_gate_kernel(
    const float* __restrict__ W, float* __restrict__ Out, int B) {
  __shared__ __align__(16) float tile[WAVES][32][ROWSTRIDE];

  const unsigned tid  = threadIdx.x;
  const unsigned lane = tid & 31u;
  const unsigned wv   = tid >> 5;
  const long rowBase  = (long)blockIdx.x * BLOCK;
  const long r        = rowBase + tid;           // this thread's row
  const bool fullTile = (rowBase + BLOCK <= B);

  float4* rowv = (float4*)&tile[wv][lane][0];
  unsigned ldsWaveBase = 0, gWaveBase = 0;

  // ---------------- gather this thread's 64 logits into q[16] ----------------
  float4 q[16];
  if (fullTile) {
    // Wave-cooperative async DMA: 32 rows = 8 KB contiguous per wave,
    // 16 issues x (32 lanes x 16 B) = 512 B contiguous per issue.
    ldsWaveBase = (unsigned)(uintptr_t)&tile[wv][0][0];
    gWaveBase   = (unsigned)((unsigned long long)(rowBase + (long)wv * 32) * (NEXPERT * 4));
    #pragma unroll
    for (int i = 0; i < 16; ++i) {
      unsigned chunk = (unsigned)i * 32u + lane;
      unsigned lds   = ldsWaveBase + (chunk >> 4) * (ROWSTRIDE * 4) + (chunk & 15u) * 16u;
      unsigned goff  = gWaveBase + chunk * 16u;
      asm volatile("global_load_async_to_lds_b128 %0, %1, %2"
                   :: "v"(lds), "v"(goff), "s"(W) : "memory");
    }
    // Only this wave consumes its tile -> no barrier, just ASYNCcnt.
    asm volatile("s_wait_asynccnt 0x0" ::: "memory");
    #pragma unroll
    for (int c = 0; c < 16; ++c) q[c] = rowv[c];
  } else if (r < B) {
    const float4* src = (const float4*)(W + (size_t)r * NEXPERT);
    #pragma unroll
    for (int c = 0; c < 16; ++c) q[c] = src[c];
  } else {
    #pragma unroll
    for (int c = 0; c < 16; ++c) q[c] = make_float4(0.f, 0.f, 0.f, 0.f);
  }

  // ---------------- exact top-8 via selection network ----------------
  float acc[8] = { q[0].x, q[0].y, q[0].z, q[0].w, q[1].x, q[1].y, q[1].z, q[1].w };
  sort8(acc);
  #pragma unroll
  for (int g = 1; g < 7; ++g) {
    float t[8] = { q[2*g].x,   q[2*g].y,   q[2*g].z,   q[2*g].w,
                   q[2*g+1].x, q[2*g+1].y, q[2*g+1].z, q[2*g+1].w };
    sort8(t);
    merge_top8(acc, t);
  }
  // Final group: only the top-8 *set* is needed (no re-sort).
  float lastg[8] = { q[14].x, q[14].y, q[14].z, q[14].w,
                     q[15].x, q[15].y, q[15].z, q[15].w };
  sort8(lastg);
  float m[8];
  #pragma unroll
  for (int k = 0; k < 8; ++k) m[k] = fmaxf(acc[k], lastg[7 - k]);

  float thresh = m[0];
  #pragma unroll
  for (int k = 1; k < 8; ++k) thresh = fminf(thresh, m[k]);   // 8th largest
  float l1 = fabsf(m[0]);
  #pragma unroll
  for (int k = 1; k < 8; ++k) l1 += fabsf(m[k]);              // L1 of kept entries
  const float inv = 1.0f / fmaxf(l1, L1_EPS);

  // ---------------- mask + renormalize (pure registers) ----------------
  #pragma unroll
  for (int c = 0; c < 16; ++c) {
    float4 v = q[c];
    v.x = (v.x >= thresh) ? v.x * inv : 0.0f;
    v.y = (v.y >= thresh) ? v.y * inv : 0.0f;
    v.z = (v.z >= thresh) ? v.z * inv : 0.0f;
    v.w = (v.w >= thresh) ? v.w * inv : 0.0f;
    q[c] = v;
  }

  // ---------------- scatter back out ----------------
  if (fullTile) {
    #pragma unroll
    for (int c = 0; c < 16; ++c) rowv[c] = q[c];
    // Make DS stores visible to the async DMA engine, then stream out coalesced.
    asm volatile("s_wait_dscnt 0x0" ::: "memory");
    #pragma unroll
    for (int i = 0; i < 16; ++i) {
      unsigned chunk = (unsigned)i * 32u + lane;
      unsigned lds   = ldsWaveBase + (chunk >> 4) * (ROWSTRIDE * 4) + (chunk & 15u) * 16u;
      unsigned goff  = gWaveBase + chunk * 16u;
      asm volatile("global_store_async_from_lds_b128 %0, %1, %2"
                   :: "v"(goff), "v"(lds), "s"(Out) : "memory");
    }
    // S_ENDPGM's implicit wait-idle covers async-store completion.
  } else if (r < B) {
    float4* dst = (float4*)(Out + (size_t)r * NEXPERT);
    #pragma unroll
    for (int c = 0; c < 16; ++c) dst[c] = q[c];
  }
}

extern "C" void kernel_launch(void* const* d_in, const int* in_sizes, int n_in,
                              void* d_out, int out_size, void* d_ws, size_t ws_size,
                              hipStream_t stream) {
  (void)n_in; (void)d_ws; (void)ws_size; (void)out_size;
  const float* W = (const float*)d_in[0];
  // d_in[1] is k (== 8); kernel is specialized for k=8 / E=64 like the reference.
  float* Out = (float*)d_out;
  int B = in_sizes[0] / NEXPERT;
  int grid = (B + BLOCK - 1) / BLOCK;
  topk_gate_kernel<<<grid, BLOCK, 0, stream>>>(W, Out, B);
}